// BloodSugarPredictor_74448963109354
// MI455X (gfx1250) — compile-verified
//
#include <hip/hip_runtime.h>
#include <hip/hip_bf16.h>

// MI455X / gfx1250 fused 4-layer LSTM stack + dense head.
// Batch-parallel persistent recurrence: 16 workgroups x 16 batch rows; all
// hidden/cell state lives in LDS; V_WMMA_F32_16X16X4_F32 does the gate GEMMs
// in full fp32 (288 recurrent steps are precision sensitive). Each wave owns
// one 16-wide slice of u and computes its i/f/g/o tiles in four accumulators,
// so the nonlinearity + cell update is register-resident.
//
// Round-2 refinements:
//  * Weights pre-transposed once into d_ws as [4U, DIN]: B operand = one
//    contiguous global_load_b64 per lane per K-chunk.
//  * Single base address pair per weight matrix; gate blocks (g*UN*DIN) and
//    K-chunks are compile-time immediates -> loads can clause + pipeline
//    instead of the load/wait(0)/wmma lock-step seen last round.
//  * Bias moved out of the accumulator init (was a v_mov splat chain on the
//    WMMA critical path) into the post-barrier elementwise phase; first WMMA
//    consumes inline-0 C.
//  * x staged via one b128 load / thread.

typedef __attribute__((ext_vector_type(2))) float v2f;
typedef __attribute__((ext_vector_type(8))) float v8f;

#define BSLICE 16     // batch rows per workgroup (one WMMA tile row)
#define TSTEPS 288
#define NTHREADS 256  // 8 waves (wave32)

__device__ __forceinline__ float sigmoidf_fast(float x) {
    return 1.0f / (1.0f + __expf(-x));
}

// ---------------------------------------------------------------------------
// Prep: transpose W[rows][cols] (row-major) -> Wt[cols][rows] in workspace.
// ---------------------------------------------------------------------------
__global__ void transpose_weights_kernel(const float* __restrict__ in,
                                         float* __restrict__ out,
                                         int rows, int cols) {
    const int idx = blockIdx.x * blockDim.x + threadIdx.x;
    if (idx < rows * cols) {
        const int k = idx / cols;   // input row   (K index)
        const int n = idx % cols;   // input col   (N index)
        out[(size_t)n * rows + k] = in[idx];
    }
}

// ---------------------------------------------------------------------------
// One LSTM layer step for a [16, UN] state slice.
//   xprev_lds : [16, DIN] inputs (LDS)
//   Wt : [4U, DIN] transposed kernel, Ut : [4U, UN] transposed recurrent
//   bias : [4U];  h_lds / c_lds : [16, UN] state (LDS), updated in place.
// Wave w (if w < UN/16) owns columns j0 = w*16 .. j0+15 and computes the four
// gate tiles at N = g*UN + j0 for g in {i,f,g,o}.
// ---------------------------------------------------------------------------
template <int DIN, int UN>
__device__ __forceinline__ void lstm_layer(const float* __restrict__ Wt,
                                           const float* __restrict__ Ut,
                                           const float* __restrict__ bias,
                                           const float* __restrict__ xprev_lds,
                                           float* __restrict__ h_lds,
                                           float* __restrict__ c_lds,
                                           int tid) {
    const int wave = tid >> 5;
    const int lane = tid & 31;
    const int ntj  = UN >> 4;            // 16-col tiles across u
    const bool active = (wave < ntj);
    const int j0   = (wave & (ntj - 1)) << 4;
    const int mn   = lane & 15;          // A row index == B col index (in tile)
    const int koff = (lane >> 4) << 1;   // K sub-offset per ISA f32 A/B layout

    v8f acc[4];

    if (active) {
#pragma unroll
        for (int g = 0; g < 4; ++g)
            acc[g] = (v8f){0.f, 0.f, 0.f, 0.f, 0.f, 0.f, 0.f, 0.f};

        // Single base pointer per matrix; gate block (g*UN*DIN) and K-chunk
        // (k0) offsets are compile-time immediates in the unrolled body.
        const float* Wb = Wt + (size_t)(j0 + mn) * DIN + koff;
        const float* Ub = Ut + (size_t)(j0 + mn) * UN + koff;
        const float* Ax = xprev_lds + mn * DIN + koff;
        const float* Ah = h_lds + mn * UN + koff;

        // x @ W contribution (K = DIN)
#pragma unroll 4
        for (int k0 = 0; k0 < DIN; k0 += 4) {
            const v2f a = *(const v2f*)(Ax + k0);
#pragma unroll
            for (int g = 0; g < 4; ++g) {
                const v2f b = *(const v2f*)(Wb + g * (UN * DIN) + k0);
                acc[g] = __builtin_amdgcn_wmma_f32_16x16x4_f32(
                    false, a, false, b, (short)0, acc[g], false, false);
            }
        }

        // h @ U contribution (K = UN)
#pragma unroll 4
        for (int k0 = 0; k0 < UN; k0 += 4) {
            const v2f a = *(const v2f*)(Ah + k0);
#pragma unroll
            for (int g = 0; g < 4; ++g) {
                const v2f b = *(const v2f*)(Ub + g * (UN * UN) + k0);
                acc[g] = __builtin_amdgcn_wmma_f32_16x16x4_f32(
                    false, a, false, b, (short)0, acc[g], false, false);
            }
        }
    }

    // All waves must finish READING h_lds before anyone writes it.
    __syncthreads();

    if (active) {
        // C/D layout: vgpr r -> M = r + (lane>=16 ? 8 : 0), N = lane & 15.
        const int mbase = (lane >> 4) << 3;
        const int j     = j0 + mn;
        const float bii = bias[0 * UN + j];
        const float bff = bias[1 * UN + j];
        const float bgg = bias[2 * UN + j];
        const float boo = bias[3 * UN + j];
#pragma unroll
        for (int r = 0; r < 8; ++r) {
            const int m = r + mbase;
            const float iv = sigmoidf_fast(acc[0][r] + bii);
            const float fv = sigmoidf_fast(acc[1][r] + bff);
            const float gv = fmaxf(acc[2][r] + bgg, 0.0f);   // cell act: relu
            const float ov = sigmoidf_fast(acc[3][r] + boo);
            const float cc = fv * c_lds[m * UN + j] + iv * gv;
            const float hh = ov * fmaxf(cc, 0.0f);           // out act: relu
            c_lds[m * UN + j] = cc;
            h_lds[m * UN + j] = hh;
        }
    }

    // New h visible to the next layer's WMMA phase.
    __syncthreads();
}

__global__ __launch_bounds__(NTHREADS) void lstm_stack_head_kernel(
    const float* __restrict__ x,                                   // [256,288,64]
    const float* __restrict__ W1t, const float* __restrict__ U1t, const float* __restrict__ b1,
    const float* __restrict__ W2t, const float* __restrict__ U2t, const float* __restrict__ b2,
    const float* __restrict__ W3t, const float* __restrict__ U3t, const float* __restrict__ b3,
    const float* __restrict__ W4t, const float* __restrict__ U4t, const float* __restrict__ b4,
    const float* __restrict__ Wf, const float* __restrict__ bf,
    const float* __restrict__ Wo, const float* __restrict__ bo,
    float* __restrict__ out) {
    __shared__ float xbuf[BSLICE * 64];
    __shared__ float h1[BSLICE * 128], c1[BSLICE * 128];
    __shared__ float h2[BSLICE * 64],  c2[BSLICE * 64];
    __shared__ float h3[BSLICE * 64],  c3[BSLICE * 64];
    __shared__ float h4[BSLICE * 32],  c4[BSLICE * 32];
    __shared__ float fcb[BSLICE * 16];

    const int tid   = threadIdx.x;
    const int bbase = blockIdx.x * BSLICE;

    // Zero-initialize recurrent state.
    for (int i = tid; i < BSLICE * 128; i += NTHREADS) { h1[i] = 0.0f; c1[i] = 0.0f; }
    for (int i = tid; i < BSLICE * 64;  i += NTHREADS) {
        h2[i] = 0.0f; c2[i] = 0.0f; h3[i] = 0.0f; c3[i] = 0.0f;
    }
    for (int i = tid; i < BSLICE * 32;  i += NTHREADS) { h4[i] = 0.0f; c4[i] = 0.0f; }
    __syncthreads();

    // Per-thread x staging slot: 256 threads cover 16 rows x 64 cols as b128s.
    const int xm = tid >> 4;          // batch row in slice
    const int xc = (tid & 15) << 2;   // starting col (multiple of 4)
    const float* xsrc = x + (size_t)(bbase + xm) * (TSTEPS * 64) + xc;

    for (int t = 0; t < TSTEPS; ++t) {
        // Stage x[bbase..bbase+15, t, :] into LDS: one b128 per thread.
        *(float4*)(xbuf + xm * 64 + xc) = *(const float4*)(xsrc + (size_t)t * 64);
        __syncthreads();

        lstm_layer<64, 128>(W1t, U1t, b1, xbuf, h1, c1, tid);
        lstm_layer<128, 64>(W2t, U2t, b2, h1,  h2, c2, tid);
        lstm_layer<64, 64>(W3t, U3t, b3, h2,  h3, c3, tid);
        lstm_layer<64, 32>(W4t, U4t, b4, h3,  h4, c4, tid);
    }

    // Dense head: relu(h4 @ Wf + bf) @ Wo + bo. Tiny: plain VALU.
    {
        const int m = tid >> 4;   // batch row in slice
        const int j = tid & 15;   // fc unit
        float s = bf[j];
#pragma unroll
        for (int k = 0; k < 32; ++k) s += h4[m * 32 + k] * Wf[k * 16 + j];
        fcb[m * 16 + j] = fmaxf(s, 0.0f);
    }
    __syncthreads();
    if (tid < BSLICE) {
        float s = bo[0];
#pragma unroll
        for (int j = 0; j < 16; ++j) s += fcb[tid * 16 + j] * Wo[j];
        out[bbase + tid] = s;
    }
}

extern "C" void kernel_launch(void* const* d_in, const int* in_sizes, int n_in,
                              void* d_out, int out_size, void* d_ws, size_t ws_size,
                              hipStream_t stream) {
    (void)in_sizes; (void)n_in; (void)out_size; (void)ws_size;
    const float* x  = (const float*)d_in[0];
    const float* W1 = (const float*)d_in[1];
    const float* U1 = (const float*)d_in[2];
    const float* b1 = (const float*)d_in[3];
    const float* W2 = (const float*)d_in[4];
    const float* U2 = (const float*)d_in[5];
    const float* b2 = (const float*)d_in[6];
    const float* W3 = (const float*)d_in[7];
    const float* U3 = (const float*)d_in[8];
    const float* b3 = (const float*)d_in[9];
    const float* W4 = (const float*)d_in[10];
    const float* U4 = (const float*)d_in[11];
    const float* b4 = (const float*)d_in[12];
    const float* Wf = (const float*)d_in[13];
    const float* bf = (const float*)d_in[14];
    const float* Wo = (const float*)d_in[15];
    const float* bo = (const float*)d_in[16];
    float* out = (float*)d_out;
    float* ws  = (float*)d_ws;

    // Workspace layout (floats): transposed weights [4U][DIN].
    float* W1t = ws;               // [512][64]
    float* U1t = W1t + 512 * 64;   // [512][128]
    float* W2t = U1t + 512 * 128;  // [256][128]
    float* U2t = W2t + 256 * 128;  // [256][64]
    float* W3t = U2t + 256 * 64;   // [256][64]
    float* U3t = W3t + 256 * 64;   // [256][64]
    float* W4t = U3t + 256 * 64;   // [128][64]
    float* U4t = W4t + 128 * 64;   // [128][32]
    // total: 192512 floats = 752.0 KiB

    auto tgrid = [](int n) { return dim3((n + NTHREADS - 1) / NTHREADS); };
    transpose_weights_kernel<<<tgrid(64 * 512),  NTHREADS, 0, stream>>>(W1, W1t, 64, 512);
    transpose_weights_kernel<<<tgrid(128 * 512), NTHREADS, 0, stream>>>(U1, U1t, 128, 512);
    transpose_weights_kernel<<<tgrid(128 * 256), NTHREADS, 0, stream>>>(W2, W2t, 128, 256);
    transpose_weights_kernel<<<tgrid(64 * 256),  NTHREADS, 0, stream>>>(U2, U2t, 64, 256);
    transpose_weights_kernel<<<tgrid(64 * 256),  NTHREADS, 0, stream>>>(W3, W3t, 64, 256);
    transpose_weights_kernel<<<tgrid(64 * 256),  NTHREADS, 0, stream>>>(U3, U3t, 64, 256);
    transpose_weights_kernel<<<tgrid(64 * 128),  NTHREADS, 0, stream>>>(W4, W4t, 64, 128);
    transpose_weights_kernel<<<tgrid(32 * 128),  NTHREADS, 0, stream>>>(U4, U4t, 32, 128);

    dim3 grid(256 / BSLICE);   // 16 workgroups, one per 16-row batch slice
    dim3 block(NTHREADS);      // 8 wave32 waves
    lstm_stack_head_kernel<<<grid, block, 0, stream>>>(
        x, W1t, U1t, b1, W2t, U2t, b2, W3t, U3t, b3, W4t, U4t, b4,
        Wf, bf, Wo, bo, out);
}